// DetectionLoss_29283087024789
// MI455X (gfx1250) — compile-verified
//
#include <hip/hip_runtime.h>

// ---------------------------------------------------------------------------
// DetectionLoss for MI455X (gfx1250, wave32).
//   preds:   (16, 8, 512, 512) f32   targets: (16, 64, 7) f32
//   out:     [total_loss, num_objects] f32
//
// Key restructuring: mask is nonzero at <=1024 of 4.19M cells, so only the
// cls channel (16 MiB) needs streaming; reg loss + cls correction are sparse
// gathers. Wave reductions use V_WMMA_F32_16X16X4_F32 with an all-ones B:
//   D = A x Ones  -> row sums; per-lane add of the 8 D regs -> half-wave sums;
//   second WMMA   -> full wave sum broadcast to every lane. Exact f32, no
//   shuffles, exercises the CDNA5 WMMA pipe.
// ---------------------------------------------------------------------------

typedef __attribute__((ext_vector_type(2))) float v2f;
typedef __attribute__((ext_vector_type(8))) float v8f;

#define B_      16
#define C_      8
#define H_      512
#define W_      512
#define HW_     (H_ * W_)           // 262144
#define CHW_    (C_ * HW_)          // 2097152
#define BHW_    (B_ * HW_)          // 4194304 cls elements
#define NF4_    (BHW_ / 4)          // 1048576 float4s of channel 0
#define HW4_    (HW_ / 4)           // 65536  float4s per batch (2^16)
#define CHW4_   (CHW_ / 4)          // 524288 = 2^19
#define NT_     64                  // targets per batch
#define GB_     512                 // streaming grid blocks
#define SCALE_  (512.0f / 80.0f)    // W / GRID_EXTENT

// Exact full-wave (32 lane) sum via two chained f32 WMMAs.
// Requires EXEC all-ones (call from convergent code, full waves).
__device__ __forceinline__ float wave_sum_wmma(float s) {
    v2f a;    a[0] = s;    a[1] = 0.0f;
    v2f ones; ones[0] = 1.0f; ones[1] = 1.0f;
    v8f c = {};
    // D[m][n] = sum_k A[m][k] : row m holds s[m] + s[m+16]
    v8f d = __builtin_amdgcn_wmma_f32_16x16x4_f32(
        false, a, false, ones, (short)0, c, false, false);
    // Lane l owns rowsums {0..7} (l<16) or {8..15} (l>=16) in its 8 D regs.
    float t = ((d[0] + d[1]) + (d[2] + d[3])) + ((d[4] + d[5]) + (d[6] + d[7]));
    v2f a2; a2[0] = t; a2[1] = 0.0f;
    // Row m of A2 = S_low (k=0, lanes 0-15) + S_high (k=2, lanes 16-31)
    // => every element of D2 is the full wave sum.
    v8f d2 = __builtin_amdgcn_wmma_f32_16x16x4_f32(
        false, a2, false, ones, (short)0, c, false, false);
    return d2[0];
}

__device__ __forceinline__ float softplus(float x) {
    // max(x,0) + log1p(exp(-|x|)) == log(1 + e^x), numerically stable
    return fmaxf(x, 0.0f) + log1pf(expf(-fabsf(x)));
}

// ---------------------------------------------------------------------------
// Kernel A: sparse terms. One block, 256 threads, 4 (b,n) pairs per thread.
//   ws[0] = sum of cls logits at masked cells   (the -x*z correction)
//   ws[1] = sum of smooth-L1 reg terms at masked cells
//   ws[2] = num_objects (unique masked cells)
// ---------------------------------------------------------------------------
__global__ void det_sparse_kernel(const float* __restrict__ preds,
                                  const float* __restrict__ targets,
                                  float* __restrict__ ws) {
    __shared__ int cells[B_ * NT_];     // 1024
    __shared__ float l0[8], l1[8], l2[8];
    const int t = threadIdx.x;

    // Phase 1: cell index for every (b, n) pair. p = b*64 + n.
    #pragma unroll
    for (int k = 0; k < 4; ++k) {
        const int p = t + k * 256;
        const float* tg = targets + p * 7;
        float fx = tg[0] * SCALE_;
        float fy = tg[1] * SCALE_;
        int gx = (int)fminf(fmaxf(fx, 0.0f), (float)(W_ - 1));
        int gy = (int)fminf(fmaxf(fy, 0.0f), (float)(H_ - 1));
        cells[p] = gy * W_ + gx;
    }
    __syncthreads();

    // Phase 2: last-write-wins winners (sequential scatter-set semantics).
    float cls_c = 0.0f, reg_s = 0.0f, cnt = 0.0f;
    #pragma unroll
    for (int k = 0; k < 4; ++k) {
        const int p = t + k * 256;
        const int b = p >> 6;
        const int cell = cells[p];
        bool win = true;
        const int end = (b << 6) | (NT_ - 1);
        for (int q = p + 1; q <= end; ++q)
            if (cells[q] == cell) { win = false; break; }
        if (win) {
            cnt += 1.0f;
            const int base = b * CHW_ + cell;
            cls_c += preds[base];                       // channel 0 logit
            const float* tg = targets + p * 7;
            #pragma unroll
            for (int c = 0; c < 7; ++c) {
                float d  = preds[base + (c + 1) * HW_] - tg[c];
                float ad = fabsf(d);
                reg_s += (ad < 1.0f) ? 0.5f * d * d : ad - 0.5f;
            }
        }
    }

    // Convergent block reduction: WMMA wave sums -> LDS -> thread 0.
    float w0 = wave_sum_wmma(cls_c);
    float w1 = wave_sum_wmma(reg_s);
    float w2 = wave_sum_wmma(cnt);
    const int lane = t & 31, wid = t >> 5;
    if (lane == 0) { l0[wid] = w0; l1[wid] = w1; l2[wid] = w2; }
    __syncthreads();
    if (t == 0) {
        float r0 = 0.f, r1 = 0.f, r2 = 0.f;
        #pragma unroll
        for (int i = 0; i < 8; ++i) { r0 += l0[i]; r1 += l1[i]; r2 += l2[i]; }
        ws[0] = r0; ws[1] = r1; ws[2] = r2;
    }
}

// ---------------------------------------------------------------------------
// Kernel B: stream channel 0 (16 MiB), accumulate softplus.
// 512 blocks x 256 threads, exactly 8 float4 per thread (no tail).
// ---------------------------------------------------------------------------
__global__ void det_cls_stream_kernel(const float* __restrict__ preds,
                                      float* __restrict__ partials) {
    __shared__ float lred[8];
    const float4* __restrict__ p4 = (const float4*)preds;
    const unsigned i = blockIdx.x * 256u + threadIdx.x;

    // Issue all 8 b128 loads up front for memory-level parallelism.
    float4 f[8];
    #pragma unroll
    for (int k = 0; k < 8; ++k) {
        const unsigned v = i + (unsigned)k * (GB_ * 256u);   // < NF4_
        const unsigned b = v >> 16;                          // / HW4_
        const unsigned r = v & (HW4_ - 1u);
        f[k] = p4[(b << 19) | r];                            // b*CHW4_ + r
    }
    float acc = 0.0f;
    #pragma unroll
    for (int k = 0; k < 8; ++k)
        acc += (softplus(f[k].x) + softplus(f[k].y)) +
               (softplus(f[k].z) + softplus(f[k].w));

    const float w = wave_sum_wmma(acc);
    const int lane = threadIdx.x & 31, wid = threadIdx.x >> 5;
    if (lane == 0) lred[wid] = w;
    __syncthreads();
    if (threadIdx.x == 0) {
        float r = 0.f;
        #pragma unroll
        for (int j = 0; j < 8; ++j) r += lred[j];
        partials[blockIdx.x] = r;                            // deterministic
    }
}

// ---------------------------------------------------------------------------
// Kernel C: fixed-order final combine. One block, 256 threads.
// ---------------------------------------------------------------------------
__global__ void det_final_kernel(const float* __restrict__ ws,
                                 float* __restrict__ out) {
    __shared__ float lred[8];
    const int t = threadIdx.x;
    const float* part = ws + 16;
    float s = part[t] + part[t + 256];                       // GB_ = 512
    const float w = wave_sum_wmma(s);
    if ((t & 31) == 0) lred[t >> 5] = w;
    __syncthreads();
    if (t == 0) {
        float sp_sum = 0.f;
        #pragma unroll
        for (int j = 0; j < 8; ++j) sp_sum += lred[j];
        const float cls_loss = (sp_sum - ws[0]) / (float)BHW_;
        const float nobj     = ws[2];
        float reg_loss = ws[1] / (nobj + 1e-6f);
        if (!(nobj > 0.0f)) reg_loss = 0.0f;
        out[0] = cls_loss + 2.0f * reg_loss;
        out[1] = nobj;
    }
}

// ---------------------------------------------------------------------------
extern "C" void kernel_launch(void* const* d_in, const int* in_sizes, int n_in,
                              void* d_out, int out_size, void* d_ws, size_t ws_size,
                              hipStream_t stream) {
    (void)in_sizes; (void)n_in; (void)out_size; (void)ws_size;
    const float* preds   = (const float*)d_in[0];
    const float* targets = (const float*)d_in[1];
    float*       out     = (float*)d_out;
    float*       ws      = (float*)d_ws;   // [0..2] sparse sums, [16..16+GB_) partials

    det_sparse_kernel    <<<1,   256, 0, stream>>>(preds, targets, ws);
    det_cls_stream_kernel<<<GB_, 256, 0, stream>>>(preds, ws + 16);
    det_final_kernel     <<<1,   256, 0, stream>>>(ws, out);
}